// ConvLRULayer_9131100471896
// MI455X (gfx1250) — compile-verified
//
#include <hip/hip_runtime.h>
#include <math.h>
#include <stdint.h>

// ---------------------------------------------------------------------------
// FFT-based 2D circular convolution for B=4,C=64,H=251,W=509 (pad 256x512).
// y = real(iDFT2(rfft2(u)*rfft2(k))) * (Hp*Wp)^{-1.5}   (ortho norm folded)
// DFT-256 = two radix-16 stages, each a 16x16 complex matmul built from
// V_WMMA_F32_16X16X4_F32 (4 real products x 4 K-chunks = 16 wmma per stage).
// u and k share one complex forward transform via z = u + i*k packing.
// Input/spectrum tiles are staged into LDS with the Tensor Data Mover
// (tensor_load_to_lds); TDM OOB-zero-fill implements the zero padding.
// ---------------------------------------------------------------------------

typedef __attribute__((ext_vector_type(2))) float v2f;
typedef __attribute__((ext_vector_type(8))) float v8f;
typedef __attribute__((ext_vector_type(4))) unsigned int u32x4;
typedef __attribute__((ext_vector_type(8))) int i32x8;
typedef __attribute__((ext_vector_type(4))) int i32x4;

#define NPLANE 256          // B*C
#define HH 251
#define WW 509
#define HP 256
#define WP 512
#define TPB 128             // 4 waves of 32
#define PI2 6.28318530717958647692f
#define OUTSCALE 2.1073424255447017e-08f   // (256*512)^-1.5 = 2^-25.5

static __device__ __forceinline__ v8f wmma4(v2f a, v2f b, v8f c) {
  // D(16x16,f32) = A(16x4,f32) * B(4x16,f32) + C
  return __builtin_amdgcn_wmma_f32_16x16x4_f32(
      /*neg_a=*/false, a, /*neg_b=*/false, b,
      /*c_mod=*/(short)0, c, /*reuse_a=*/false, /*reuse_b=*/false);
}

// Forward 16-point DFT matrix element: W16[r][c] = exp(-2*pi*i*(r*c mod 16)/16)
static __device__ __forceinline__ void w16f(int r, int c, float& re, float& im) {
  float ang = -(PI2 / 16.0f) * (float)((r * c) & 15);
  re = __cosf(ang);
  im = __sinf(ang);
}

// ---------------------------------------------------------------------------
// TDM: issue one 2D tensor_load_to_lds.  D# groups packed per ISA ch.8:
//   group0: count=1 | lds_addr | global_addr[56:0] | type=2
//   group1: wg_mask=0, data_size, tensor_dim0/1, tile_dim0/1, dim0 stride
// OOB rows/cols (index >= tensor_dim) read as zero -> hardware zero-padding.
// Must be executed by a single wave; completion via s_wait_tensorcnt.
// ---------------------------------------------------------------------------
static __device__ __forceinline__ void tdm_load_2d(
    unsigned lds_off, unsigned long long gaddr, unsigned data_size_log,
    unsigned tensor_d0, unsigned tensor_d1, unsigned tile_d0, unsigned tile_d1,
    unsigned long long stride0_elems) {
  u32x4 g0;
  g0.x = 1u;                                                 // count=1 (valid)
  g0.y = lds_off;                                            // LDS byte offset
  g0.z = (unsigned)(gaddr & 0xffffffffull);
  g0.w = (unsigned)((gaddr >> 32) & 0x1ffffffull) | 0x80000000u;  // type=2
  i32x8 g1;
  g1[0] = (int)(data_size_log << 16);                        // mask=0|data_size
  g1[1] = (int)((tensor_d0 & 0xffffu) << 16);                // dim0 lo16
  g1[2] = (int)((tensor_d0 >> 16) | ((tensor_d1 & 0xffffu) << 16));
  g1[3] = (int)((tensor_d1 >> 16) | ((tile_d0 & 0xffffu) << 16));
  g1[4] = (int)(tile_d1 & 0xffffu);                          // tile_dim1|2=0
  g1[5] = (int)(stride0_elems & 0xffffffffull);              // dim0 stride lo
  g1[6] = (int)((stride0_elems >> 32) & 0xffffull);          // hi | d1str lo=0
  g1[7] = 0;
  i32x4 z4 = {0, 0, 0, 0};
#if __clang_major__ >= 23
  i32x8 z8 = {};
  __builtin_amdgcn_tensor_load_to_lds(g0, g1, z4, z4, z8, 0);
#else
  __builtin_amdgcn_tensor_load_to_lds(g0, g1, z4, z4, 0);
#endif
}

static __device__ __forceinline__ unsigned lds_off_of(const void* p) {
  // flat shared address: addr[31:0] is the LDS byte offset (aperture rule)
  return (unsigned)(uintptr_t)p;
}

// ---------------------------------------------------------------------------
// In-place unnormalized DFT-256 of 16 interleaved sequences held in LDS as
// x[pos*16 + seq].  sgn=+1 -> exp(-i..) forward, sgn=-1 -> exp(+i..) inverse
// (no 1/N).  Radix-16 x radix-16 via WMMA.  Each wave owns 4 sequences; all
// LDS traffic is sequence-local -> no intra-function barriers needed.
// Wave-uniform control flow (EXEC = all ones for every WMMA).
// ---------------------------------------------------------------------------
static __device__ __forceinline__ void fft256_b16(float* xr, float* xi, float sgn) {
  const int tid  = threadIdx.x;
  const int lane = tid & 31;
  const int wid  = tid >> 5;
  const int m    = lane & 15;   // row (A) / col (B,D) index
  const int hi   = lane >> 4;   // lane half selects K pair / M half

  for (int ss = 0; ss < 4; ++ss) {
    const int s = (wid << 2) | ss;

    // ---- stage 1:  Y[k1][h2] = sum_h1 Wc[k1][h1] * X[h1*16+h2] ----
    v8f p1 = {}, p2 = {}, p3 = {}, p4 = {};
#pragma unroll
    for (int kb = 0; kb < 4; ++kb) {
      const int k0 = kb * 4 + hi * 2;
      float w0r, w0i, w1r, w1i;
      w16f(m, k0,     w0r, w0i);
      w16f(m, k0 + 1, w1r, w1i);
      v2f awr = { w0r, w1r };
      v2f awi = { w0i, w1i };
      v2f bxr = { xr[((k0)     * 16 + m) * 16 + s],
                  xr[((k0 + 1) * 16 + m) * 16 + s] };
      v2f bxi = { xi[((k0)     * 16 + m) * 16 + s],
                  xi[((k0 + 1) * 16 + m) * 16 + s] };
      p1 = wmma4(awr, bxr, p1);   // Wr*Xr
      p2 = wmma4(awi, bxi, p2);   // Wi*Xi
      p3 = wmma4(awi, bxr, p3);   // Wi*Xr
      p4 = wmma4(awr, bxi, p4);   // Wr*Xi
    }
    v8f yr = p1 - sgn * p2;
    v8f yi = sgn * p3 + p4;

    // twiddle T[k1][h2] = exp(-sgn*2*pi*i*k1*h2/256), store row-major [k1][h2]
#pragma unroll
    for (int v = 0; v < 8; ++v) {
      const int k1 = v + 8 * hi;            // D layout: M = v + 8*(lane>=16)
      const float ang = -sgn * (PI2 / 256.0f) * (float)(k1 * m);
      float sn, cs;
      __sincosf(ang, &sn, &cs);
      float re = yr[v] * cs - yi[v] * sn;
      float im = yr[v] * sn + yi[v] * cs;
      xr[(k1 * 16 + m) * 16 + s] = re;
      xi[(k1 * 16 + m) * 16 + s] = im;
    }

    // ---- stage 2:  Z[k1][k2] = sum_h2 Yt[k1][h2] * Wc[h2][k2] ----
    v8f q1 = {}, q2 = {}, q3 = {}, q4 = {};
#pragma unroll
    for (int kb = 0; kb < 4; ++kb) {
      const int k0 = kb * 4 + hi * 2;
      v2f aar = { xr[(m * 16 + k0)     * 16 + s],
                  xr[(m * 16 + k0 + 1) * 16 + s] };
      v2f aai = { xi[(m * 16 + k0)     * 16 + s],
                  xi[(m * 16 + k0 + 1) * 16 + s] };
      float w0r, w0i, w1r, w1i;
      w16f(k0,     m, w0r, w0i);
      w16f(k0 + 1, m, w1r, w1i);
      v2f bwr = { w0r, w1r };
      v2f bwi = { w0i, w1i };
      q1 = wmma4(aar, bwr, q1);   // Ar*Wr
      q2 = wmma4(aai, bwi, q2);   // Ai*Wi
      q3 = wmma4(aar, bwi, q3);   // Ar*Wi
      q4 = wmma4(aai, bwr, q4);   // Ai*Wr
    }
    v8f zr = q1 - sgn * q2;
    v8f zi = sgn * q3 + q4;

    // output index k = k2*16 + k1 (digit reversal) -> natural order store
#pragma unroll
    for (int v = 0; v < 8; ++v) {
      const int k1  = v + 8 * hi;
      const int pos = m * 16 + k1;          // k2 = m
      xr[pos * 16 + s] = zr[v];
      xi[pos * 16 + s] = zi[v];
    }
  }
}

// ---------------------------------------------------------------------------
// Kernel A: forward row FFT-512 of z = u + i*k (zero-padded 251x509 -> 256x512)
// 16 rows per block.  512 = 2 x 256 DIT: even/odd FFT-256 + combine.
// Input tiles staged via TDM; TDM OOB zero-fill performs the padding.
// ---------------------------------------------------------------------------
extern "C" __global__ void __launch_bounds__(TPB)
fftconv_rows_fwd(const float* __restrict__ u, const float* __restrict__ k,
                 float2* __restrict__ Z0) {
  extern __shared__ float sm[];
  float* er  = sm;            // [256][16]
  float* ei  = er  + 4096;
  float* orr = ei  + 4096;
  float* oi  = orr + 4096;    // 64 KB working set
  float* su  = oi  + 4096;    // [16][512] staged u tile (32 KB)
  float* sk  = su  + 8192;    // [16][512] staged k tile (32 KB)

  const int tid = threadIdx.x;
  const int p   = blockIdx.y;
  const int r0  = blockIdx.x * 16;
  const float* up = u + (size_t)p * HH * WW;
  const float* kp = k + (size_t)p * HH * WW;

  if (tid < 32) {  // one wave issues both TDM tile loads
    const unsigned rows_left = (unsigned)(HH - r0);   // r0 <= 240 < 251
    tdm_load_2d(lds_off_of(su), (unsigned long long)(uintptr_t)(up + (size_t)r0 * WW),
                /*4B*/2u, /*t_d0=*/WW, /*t_d1=*/rows_left,
                /*tile_d0=*/512u, /*tile_d1=*/16u, /*stride0=*/(unsigned long long)WW);
    tdm_load_2d(lds_off_of(sk), (unsigned long long)(uintptr_t)(kp + (size_t)r0 * WW),
                2u, WW, rows_left, 512u, 16u, (unsigned long long)WW);
    __builtin_amdgcn_s_wait_tensorcnt(0);
  }
  __syncthreads();

  // scramble staged tiles into even/odd interleaved-sequence layout
  for (int idx = tid; idx < 16 * 512; idx += TPB) {
    int rr = idx >> 9, col = idx & 511;
    float uv = su[rr * 512 + col];
    float kv = sk[rr * 512 + col];
    int q = (col >> 1) * 16 + rr;
    if (col & 1) { orr[q] = uv; oi[q] = kv; }
    else         { er[q]  = uv; ei[q]  = kv; }
  }
  __syncthreads();

  fft256_b16(er,  ei,  1.0f);
  fft256_b16(orr, oi,  1.0f);
  __syncthreads();

  float2* Zp = Z0 + (size_t)p * HP * WP;
  for (int idx = tid; idx < 16 * 256; idx += TPB) {
    int rr = idx >> 8, kk = idx & 255;
    int q = kk * 16 + rr;
    float Er = er[q], Ei = ei[q], Or = orr[q], Oi = oi[q];
    float ang = -(PI2 / 512.0f) * (float)kk;
    float sn, cs;
    __sincosf(ang, &sn, &cs);
    float tr = Or * cs - Oi * sn;
    float ti = Or * sn + Oi * cs;
    int row = r0 + rr;
    Zp[(size_t)row * WP + kk]       = make_float2(Er + tr, Ei + ti);
    Zp[(size_t)row * WP + kk + 256] = make_float2(Er - tr, Ei - ti);
  }
}

// ---------------------------------------------------------------------------
// Kernel B: column FFT-256, unpack U*K from packed z-transform, inverse
// column FFT-256.  16 mirror-paired columns per block (33 blocks/plane with
// benign same-value overlap), ping-pong Z0 -> Z1 to avoid cross-block races.
//   P(h,w) = -i * (Z(h,w)^2 - conj(Z(-h,-w))^2) / 4
// ---------------------------------------------------------------------------
extern "C" __global__ void __launch_bounds__(TPB)
fftconv_cols_mul(const float2* __restrict__ Z0, float2* __restrict__ Z1) {
  extern __shared__ float sm[];
  float* xr = sm;             // [256][16]
  float* xi = xr + 4096;
  float* pr = xi + 4096;
  float* pi = pr + 4096;      // total 64 KB

  const int tid = threadIdx.x;
  const int p   = blockIdx.y;
  const int g   = blockIdx.x;             // 0..32
  const float2* Zp = Z0 + (size_t)p * HP * WP;
  float2*       Op = Z1 + (size_t)p * HP * WP;

  for (int idx = tid; idx < 256 * 16; idx += TPB) {
    int h = idx >> 4, s = idx & 15;
    int c = (s < 8) ? (8 * g + s) : ((512 - 8 * g - 15 + s) & 511);
    __builtin_prefetch(&Zp[(size_t)((h + 32) & 255) * WP + c], 0, 1);
    float2 z = Zp[(size_t)h * WP + c];
    xr[idx] = z.x;
    xi[idx] = z.y;
  }
  __syncthreads();

  fft256_b16(xr, xi, 1.0f);               // forward column transform
  __syncthreads();

  for (int idx = tid; idx < 256 * 16; idx += TPB) {
    int h = idx >> 4, s = idx & 15;
    int hm = (256 - h) & 255, sm2 = 15 - s;   // mirror bin (-h, -w)
    float a  = xr[idx],           b  = xi[idx];
    float cr = xr[hm * 16 + sm2], ci = xi[hm * 16 + sm2];
    float dr = a * a - b * b - cr * cr + ci * ci;
    float di = 2.0f * (a * b + cr * ci);
    pr[idx] =  0.25f * di;
    pi[idx] = -0.25f * dr;
  }
  __syncthreads();

  fft256_b16(pr, pi, -1.0f);              // inverse column transform (unnorm)
  __syncthreads();

  for (int idx = tid; idx < 256 * 16; idx += TPB) {
    int h = idx >> 4, s = idx & 15;
    int c = (s < 8) ? (8 * g + s) : ((512 - 8 * g - 15 + s) & 511);
    Op[(size_t)h * WP + c] = make_float2(pr[idx], pi[idx]);
  }
}

// ---------------------------------------------------------------------------
// Kernel C: inverse row FFT-512 (unnormalized, sgn=-1), take real part,
// scale by (Hp*Wp)^{-1.5}, crop to 251x509.  Spectrum tile staged via TDM.
// ---------------------------------------------------------------------------
extern "C" __global__ void __launch_bounds__(TPB)
fftconv_rows_inv(const float2* __restrict__ Z1, float* __restrict__ out) {
  extern __shared__ float sm[];
  float*  er  = sm;
  float*  ei  = er  + 4096;
  float*  orr = ei  + 4096;
  float*  oi  = orr + 4096;                 // 64 KB working set
  float2* stg = (float2*)(oi + 4096);       // [16][512] complex tile (64 KB)

  const int tid = threadIdx.x;
  const int p   = blockIdx.y;
  const int r0  = blockIdx.x * 16;
  const float2* Zp = Z1 + (size_t)p * HP * WP;

  if (tid < 32) {  // one wave issues the TDM tile load (8B elements)
    tdm_load_2d(lds_off_of(stg),
                (unsigned long long)(uintptr_t)(Zp + (size_t)r0 * WP),
                /*8B*/3u, /*t_d0=*/512u, /*t_d1=*/16u,
                /*tile_d0=*/512u, /*tile_d1=*/16u, /*stride0=*/512ull);
    __builtin_amdgcn_s_wait_tensorcnt(0);
  }
  __syncthreads();

  for (int idx = tid; idx < 16 * 512; idx += TPB) {
    int rr = idx >> 9, kk = idx & 511;
    float2 z = stg[rr * 512 + kk];
    int q = (kk >> 1) * 16 + rr;
    if (kk & 1) { orr[q] = z.x; oi[q] = z.y; }
    else        { er[q]  = z.x; ei[q]  = z.y; }
  }
  __syncthreads();

  fft256_b16(er,  ei,  -1.0f);
  fft256_b16(orr, oi,  -1.0f);
  __syncthreads();

  float* op = out + (size_t)p * HH * WW;
  for (int idx = tid; idx < 16 * 256; idx += TPB) {
    int rr = idx >> 8, n = idx & 255;
    int row = r0 + rr;
    int q = n * 16 + rr;
    float Er = er[q], Ei = ei[q], Or = orr[q], Oi = oi[q];
    float ang = (PI2 / 512.0f) * (float)n;   // -sgn * 2*pi*n/512, sgn = -1
    float sn, cs;
    __sincosf(ang, &sn, &cs);
    float tr = Or * cs - Oi * sn;
    float y0 = (Er + tr) * OUTSCALE;        // sample n
    float y1 = (Er - tr) * OUTSCALE;        // sample n + 256
    if (row < HH) {
      if (n < WW)       op[(size_t)row * WW + n]       = y0;
      if (n + 256 < WW) op[(size_t)row * WW + n + 256] = y1;
    }
  }
}

// ---------------------------------------------------------------------------
extern "C" void kernel_launch(void* const* d_in, const int* in_sizes, int n_in,
                              void* d_out, int out_size, void* d_ws, size_t ws_size,
                              hipStream_t stream) {
  (void)in_sizes; (void)n_in; (void)out_size; (void)ws_size;
  const float* u = (const float*)d_in[0];
  const float* k = (const float*)d_in[1];
  float*     out = (float*)d_out;

  // workspace: two complex spectrum buffers, 256 planes x 256 x 512 each
  float2* Z0 = (float2*)d_ws;
  float2* Z1 = Z0 + (size_t)NPLANE * HP * WP;

  const size_t shmem_work  = 4 * 4096 * sizeof(float);          // 64 KB
  const size_t shmem_stage = shmem_work + 16 * 512 * 8;         // +64 KB TDM tile
  dim3 blk(TPB);

  fftconv_rows_fwd<<<dim3(16, NPLANE), blk, shmem_stage, stream>>>(u, k, Z0);
  fftconv_cols_mul<<<dim3(33, NPLANE), blk, shmem_work,  stream>>>(Z0, Z1);
  fftconv_rows_inv<<<dim3(16, NPLANE), blk, shmem_stage, stream>>>(Z1, out);
}